// NodeMemory_23450521436436
// MI455X (gfx1250) — compile-verified
//
#include <hip/hip_runtime.h>

typedef __attribute__((ext_vector_type(2))) float v2f;
typedef __attribute__((ext_vector_type(4))) float v4f;
typedef __attribute__((ext_vector_type(8))) float v8f;

#define H 128
#define DMSG 256
#define TM 16            // rows (updated nodes) per block
#define THREADS 256      // 8 wave32
#define XSTR 260         // 256 + 4 pad: bank = (4*row + k) & 63, conflict-free
#define HSTR 132         // 128 + 4 pad
#define GSTR 260         // 256 + 4 pad for the combined r/z gate buffer

__device__ __forceinline__ float fast_sigmoid(float x) {
    return 1.0f / (1.0f + __expf(-x));
}
__device__ __forceinline__ float fast_tanh(float x) {
    // overflow-safe: x->+inf => 1, x->-inf => -1
    float e = __expf(2.0f * x);
    return 1.0f - 2.0f / (e + 1.0f);
}

__global__ __launch_bounds__(THREADS)
void gru_scatter_kernel(const float* __restrict__ memory,
                        const int*   __restrict__ node_ids,
                        const float* __restrict__ messages,
                        const float* __restrict__ W_ih,
                        const float* __restrict__ W_hh,
                        const float* __restrict__ b_ih,
                        const float* __restrict__ b_hh,
                        float* __restrict__ out,
                        int M) {
    __shared__ float lx[TM * XSTR];      // message tile  [16 x 256]
    __shared__ float lh[TM * HSTR];      // gathered h    [16 x 128]
    __shared__ float lg_rz[TM * GSTR];   // gi+gh for r,z gates [16 x 256]
    __shared__ float lg_in[TM * HSTR];   // gi for n gate [16 x 128]
    __shared__ float lg_hn[TM * HSTR];   // gh for n gate [16 x 128]
    __shared__ int   lid[TM];

    const int tid = threadIdx.x;
    const int m0  = blockIdx.x * TM;

    if (tid < TM) {
        int r = m0 + tid;
        if (r >= M) r = M - 1;
        lid[tid] = node_ids[r];
    }

    // ---- stage messages tile: 16 rows x 256 f32 = 1024 float4 ----
#pragma unroll
    for (int i = 0; i < 4; ++i) {
        int idx = i * THREADS + tid;          // float4 slot
        int r   = idx >> 6;                   // row 0..15
        int c4  = (idx & 63) << 2;            // col 0..252
        int rg  = m0 + r; if (rg >= M) rg = M - 1;
        v4f v = *(const v4f*)(messages + (size_t)rg * DMSG + c4);
        *(v4f*)(lx + r * XSTR + c4) = v;
    }
    __syncthreads();   // lid ready

    // ---- gather h tile: 16 rows x 128 f32 = 512 float4 ----
#pragma unroll
    for (int i = 0; i < 2; ++i) {
        int idx = i * THREADS + tid;
        int r   = idx >> 5;
        int c4  = (idx & 31) << 2;
        v4f v = *(const v4f*)(memory + (size_t)lid[r] * H + c4);
        *(v4f*)(lh + r * HSTR + c4) = v;
    }
    __syncthreads();

    // ---- WMMA phase: each wave owns 3 of the 24 N-tiles (384 gate cols) ----
    const int wave = tid >> 5;
    const int lane = tid & 31;
    const int mrow = lane & 15;      // A: M row / B: N col held by this lane
    const int hi   = lane >> 4;      // 0: K0/K1, 1: K2/K3
    const int nt0  = wave * 3;

    v8f ci[3], ch[3];
    const float* wip[3];
    const float* whp[3];
#pragma unroll
    for (int t = 0; t < 3; ++t) {
        int nb = (nt0 + t) * 16;
        float bi = b_ih[nb + mrow];
        float bh = b_hh[nb + mrow];
#pragma unroll
        for (int v = 0; v < 8; ++v) { ci[t][v] = bi; ch[t][v] = bh; }
        wip[t] = W_ih + (size_t)(nb + mrow) * DMSG + 2 * hi;   // B[k][n] = W[n][k]
        whp[t] = W_hh + (size_t)(nb + mrow) * H    + 2 * hi;
    }

    // gi += X @ W_ih^T  (K = 256, 64 WMMA steps per N-tile)
#pragma unroll 4
    for (int k0 = 0; k0 < DMSG; k0 += 4) {
        v2f a = *(const v2f*)(lx + mrow * XSTR + k0 + 2 * hi);
#pragma unroll
        for (int t = 0; t < 3; ++t) {
            v2f b = *(const v2f*)(wip[t] + k0);
            ci[t] = __builtin_amdgcn_wmma_f32_16x16x4_f32(
                false, a, false, b, (short)0, ci[t], false, false);
        }
    }
    // gh += H @ W_hh^T  (K = 128, 32 WMMA steps per N-tile)
#pragma unroll 4
    for (int k0 = 0; k0 < H; k0 += 4) {
        v2f a = *(const v2f*)(lh + mrow * HSTR + k0 + 2 * hi);
#pragma unroll
        for (int t = 0; t < 3; ++t) {
            v2f b = *(const v2f*)(whp[t] + k0);
            ch[t] = __builtin_amdgcn_wmma_f32_16x16x4_f32(
                false, a, false, b, (short)0, ch[t], false, false);
        }
    }

    // ---- write gate tiles to LDS ----
    // C/D layout: element v of lane L holds (m = v + 8*(L>>4), n = L&15)
#pragma unroll
    for (int t = 0; t < 3; ++t) {
        int nb = (nt0 + t) * 16;
        if (nb < 256) {            // r/z sections: only the sum is needed
#pragma unroll
            for (int v = 0; v < 8; ++v) {
                int m = v + 8 * hi;
                lg_rz[m * GSTR + nb + mrow] = ci[t][v] + ch[t][v];
            }
        } else {                   // n section: keep gi_n and gh_n separate
            int cb = nb - 256;
#pragma unroll
            for (int v = 0; v < 8; ++v) {
                int m = v + 8 * hi;
                lg_in[m * HSTR + cb + mrow] = ci[t][v];
                lg_hn[m * HSTR + cb + mrow] = ch[t][v];
            }
        }
    }
    __syncthreads();

    // ---- elementwise GRU combine + scatter: 16 x 128 outputs ----
#pragma unroll
    for (int i = 0; i < 8; ++i) {
        int idx = i * THREADS + tid;       // 0..2047
        int m   = idx >> 7;
        int c   = idx & 127;
        float r = fast_sigmoid(lg_rz[m * GSTR + c]);
        float z = fast_sigmoid(lg_rz[m * GSTR + 128 + c]);
        float n = fast_tanh(lg_in[m * HSTR + c] + r * lg_hn[m * HSTR + c]);
        float h = lh[m * HSTR + c];
        float hn = (1.0f - z) * n + z * h;
        if (m0 + m < M)
            out[(size_t)lid[m] * H + c] = hn;
    }
}

extern "C" void kernel_launch(void* const* d_in, const int* in_sizes, int n_in,
                              void* d_out, int out_size, void* d_ws, size_t ws_size,
                              hipStream_t stream) {
    const float* memory   = (const float*)d_in[0];   // [1M, 128]
    const int*   node_ids = (const int*)  d_in[1];   // [131072]
    const float* messages = (const float*)d_in[2];   // [131072, 256]
    const float* W_ih     = (const float*)d_in[3];   // [384, 256]
    const float* W_hh     = (const float*)d_in[4];   // [384, 128]
    const float* b_ih     = (const float*)d_in[5];   // [384]
    const float* b_hh     = (const float*)d_in[6];   // [384]
    float* out = (float*)d_out;

    // 1) bulk copy of the full memory table into the output (stream-ordered,
    //    graph-capture legal); updated rows are overwritten by the kernel.
    size_t mem_bytes = (size_t)in_sizes[0] * sizeof(float);
    hipMemcpyAsync(out, memory, mem_bytes, hipMemcpyDeviceToDevice, stream);

    // 2) gather -> GRU (WMMA f32) -> scatter
    int M = in_sizes[1];
    int blocks = (M + TM - 1) / TM;          // 8192 for M = 131072
    gru_scatter_kernel<<<blocks, THREADS, 0, stream>>>(
        memory, node_ids, messages, W_ih, W_hh, b_ih, b_hh, out, M);
}